// GKAN_Nodes_70609262346476
// MI455X (gfx1250) — compile-verified
//
#include <hip/hip_runtime.h>
#include <hip/hip_bf16.h>
#include <math.h>

typedef __attribute__((ext_vector_type(2))) float v2f;
typedef __attribute__((ext_vector_type(8))) float v8f;

#define F_IN   128
#define GPTS   12          // grid points per feature (GRID_SIZE + 2*SPLINE_ORDER + 1)
#define KEXP   (F_IN * 9)  // 1152 expanded K per 128-feature block

// ---------------------------------------------------------------- utilities
__global__ void fill_f32(float* p, float v, size_t n) {
  size_t i = (size_t)blockIdx.x * blockDim.x + threadIdx.x;
  if (i < n) p[i] = v;
}

__global__ void copy_x_to_xc(const float* __restrict__ x, float* __restrict__ xc, int N) {
  size_t i = (size_t)blockIdx.x * blockDim.x + threadIdx.x;
  if (i >= (size_t)N * F_IN) return;
  int n = (int)(i >> 7), c = (int)(i & 127);
  xc[(size_t)n * 512 + c] = x[i];
}

// ---------------------------------------------------------------- degree / dinv
__global__ void deg_count(const int* __restrict__ ei, float* deg, int E) {
  int e = blockIdx.x * blockDim.x + threadIdx.x;
  if (e < E) atomicAdd(&deg[ei[E + e]], 1.0f);   // dst = edge_index[1][e]
}

__global__ void make_dinv(const float* __restrict__ deg, float* dinv, int N) {
  int n = blockIdx.x * blockDim.x + threadIdx.x;
  if (n < N) dinv[n] = rsqrtf(deg[n] + 1.0f);    // +1 self loop, always >= 1
}

// ---------------------------------------------------------------- fused weight prep
// W'[o][f*9+0] = base_w[o][f]; W'[o][f*9+1+j] = spline_w[o][f][j] * scaler[o][f]
__global__ void prep_w(const float* __restrict__ bw, const float* __restrict__ sw,
                       const float* __restrict__ sc, float* __restrict__ W,
                       int O, int Opad, int Fin) {
  int idx = blockIdx.x * blockDim.x + threadIdx.x;
  if (idx >= Opad * Fin) return;
  int o = idx / Fin, f = idx % Fin;
  float* dst = W + (size_t)o * ((size_t)Fin * 9) + (size_t)f * 9;
  if (o < O) {
    size_t of = (size_t)o * Fin + f;
    float s = sc[of];
    dst[0] = bw[of];
#pragma unroll
    for (int j = 0; j < 8; ++j) dst[1 + j] = sw[of * 8 + j] * s;
  } else {
#pragma unroll
    for (int j = 0; j < 9; ++j) dst[j] = 0.0f;
  }
}

// ---------------------------------------------------------------- fused KAN GEMM (WMMA f32)
// C(N x O) (+)= Phi(X_block)(N x 1152) @ W'(O x 1152)^T
// Phi expands each feature f into [silu(x), B0..B7] computed in LDS per chunk.
__global__ void kan_gemm(const float* __restrict__ X, int x_stride, int x_col_off,
                         const float* __restrict__ grid,     // 12 floats per feature
                         const float* __restrict__ W, int w_stride,
                         float* __restrict__ Cout, int c_stride,
                         int N, int O, int accumulate) {
  __shared__ float A_sm[64 * 37];   // 64 nodes x 36 expanded cols (stride 37)
  __shared__ float B_sm[64 * 37];   // 64 outputs x 36 k-cols

  const int tid  = threadIdx.x;
  const int wave = tid >> 5;
  const int lane = tid & 31;
  const int lo   = lane & 15;
  const int hi   = lane >> 4;

  const int m_base = blockIdx.x * 64;
  const int o_base = blockIdx.y * 64;

  const int ms  = wave >> 1;            // 4 subtile rows
  const int os0 = (wave * 2) & 3;       // 4 subtile cols, 2 per wave
  const int os1 = os0 + 1;

  v8f acc0 = {0.f,0.f,0.f,0.f,0.f,0.f,0.f,0.f};
  v8f acc1 = {0.f,0.f,0.f,0.f,0.f,0.f,0.f,0.f};

  const int n_l = tid >> 2;             // node within tile for A staging
  const int f_l = tid & 3;              // feature within chunk

  for (int ch = 0; ch < 32; ++ch) {     // 32 chunks x 4 features = 128 features
    // ---- stage A: expand 64 nodes x 4 features -> 36 cols
    {
      int node = m_base + n_l;
      int kf   = ch * 4 + f_l;          // feature index within 128-block
      float out9[9];
      if (node < N) {
        float xv = X[(size_t)node * x_stride + x_col_off + kf];
        const float* g = grid + kf * GPTS;
        float b[11];
#pragma unroll
        for (int t = 0; t < 11; ++t)
          b[t] = (xv >= g[t] && xv < g[t + 1]) ? 1.0f : 0.0f;
#pragma unroll
        for (int k = 1; k <= 3; ++k) {
#pragma unroll
          for (int t = 0; t < 11 - 3; ++t) {      // unified bound; extras guarded below
            if (t < 11 - k) {
              float left  = (xv - g[t]) / (g[t + k] - g[t]);
              float right = (g[t + k + 1] - xv) / (g[t + k + 1] - g[t + 1]);
              b[t] = left * b[t] + right * b[t + 1];
            }
          }
          // handle tail entries for k=1,2 (t in [8, 11-k))
#pragma unroll
          for (int t = 8; t < 10; ++t) {
            if (t < 11 - k) {
              float left  = (xv - g[t]) / (g[t + k] - g[t]);
              float right = (g[t + k + 1] - xv) / (g[t + k + 1] - g[t + 1]);
              b[t] = left * b[t] + right * b[t + 1];
            }
          }
        }
        out9[0] = xv / (1.0f + expf(-xv));        // SiLU
#pragma unroll
        for (int j = 0; j < 8; ++j) out9[1 + j] = b[j];
      } else {
#pragma unroll
        for (int j = 0; j < 9; ++j) out9[j] = 0.0f;
      }
      float* dst = &A_sm[n_l * 37 + f_l * 9];
#pragma unroll
      for (int j = 0; j < 9; ++j) dst[j] = out9[j];
    }
    // ---- stage B: W' slice, 64 outputs x 36 cols (2304 = 9 * 256)
    {
#pragma unroll
      for (int i = 0; i < 9; ++i) {
        int idx = tid + i * 256;
        int row = idx / 36, col = idx - row * 36;
        int o = o_base + row;
        float v = 0.0f;
        if (o < O) v = W[(size_t)o * w_stride + ch * 36 + col];
        B_sm[row * 37 + col] = v;
      }
    }
    __syncthreads();

    const float* Arow  = &A_sm[(ms  * 16 + lo) * 37 + 2 * hi];
    const float* Brow0 = &B_sm[(os0 * 16 + lo) * 37 + 2 * hi];
    const float* Brow1 = &B_sm[(os1 * 16 + lo) * 37 + 2 * hi];
#pragma unroll
    for (int kk = 0; kk < 9; ++kk) {
      v2f a;  a.x  = Arow[kk * 4];  a.y  = Arow[kk * 4 + 1];
      v2f b0; b0.x = Brow0[kk * 4]; b0.y = Brow0[kk * 4 + 1];
      v2f b1; b1.x = Brow1[kk * 4]; b1.y = Brow1[kk * 4 + 1];
      acc0 = __builtin_amdgcn_wmma_f32_16x16x4_f32(false, a, false, b0,
                                                   (short)0, acc0, false, false);
      acc1 = __builtin_amdgcn_wmma_f32_16x16x4_f32(false, a, false, b1,
                                                   (short)0, acc1, false, false);
    }
    __syncthreads();
  }

  // ---- store 16x16 subtiles (C/D layout: VGPR r -> M=r (+8 for hi half), lane -> N)
  const int col0 = o_base + os0 * 16 + lo;
  const int col1 = o_base + os1 * 16 + lo;
#pragma unroll
  for (int r = 0; r < 8; ++r) {
    int row = m_base + ms * 16 + r + 8 * hi;
    if (row < N) {
      if (col0 < O) {
        size_t off = (size_t)row * c_stride + col0;
        Cout[off] = acc0[r] + (accumulate ? Cout[off] : 0.0f);
      }
      if (col1 < O) {
        size_t off = (size_t)row * c_stride + col1;
        Cout[off] = acc1[r] + (accumulate ? Cout[off] : 0.0f);
      }
    }
  }
}

// ---------------------------------------------------------------- aggregation
// hagg[n][c] = hlin[n][c] * dinv[n]^2 + bias[c]   (self loop + bias init)
__global__ void agg_init(const float* __restrict__ h, const float* __restrict__ dinv,
                         const float* __restrict__ bias, float* __restrict__ hagg, int N) {
  size_t i = (size_t)blockIdx.x * blockDim.x + threadIdx.x;
  if (i >= (size_t)N * 128) return;
  int n = (int)(i >> 7), c = (int)(i & 127);
  float w = dinv[n];
  hagg[i] = h[i] * w * w + bias[c];
}

// one wave per edge; lane handles 4 channels via float4 gather + scalar atomics
__global__ void agg_edges(const int* __restrict__ ei, const float* __restrict__ h,
                          const float* __restrict__ dinv, float* __restrict__ hagg, int E) {
  int gid = blockIdx.x * blockDim.x + threadIdx.x;
  int e = gid >> 5;
  if (e >= E) return;
  int lane = gid & 31;
  int s = ei[e];
  int d = ei[E + e];
  float w = dinv[s] * dinv[d];
  const float4* hs = (const float4*)(h + (size_t)s * 128);
  float4 v = hs[lane];
  float* hd = hagg + (size_t)d * 128 + lane * 4;
  atomicAdd(hd + 0, v.x * w);
  atomicAdd(hd + 1, v.y * w);
  atomicAdd(hd + 2, v.z * w);
  atomicAdd(hd + 3, v.w * w);
}

// ---------------------------------------------------------------- batch norm
__global__ void bn_reduce(const float* __restrict__ h, float* sum, float* sumsq, int N) {
  __shared__ float s1[256], s2[256];
  int c = threadIdx.x & 127;
  int g = threadIdx.x >> 7;
  int base = blockIdx.x * 128;
  int end = base + 128; if (end > N) end = N;
  float a = 0.f, b = 0.f;
  for (int r = base + g; r < end; r += 2) {
    float v = h[(size_t)r * 128 + c];
    a += v; b += v * v;
  }
  s1[threadIdx.x] = a; s2[threadIdx.x] = b;
  __syncthreads();
  if (g == 0) {
    atomicAdd(&sum[c],   a + s1[threadIdx.x + 128]);
    atomicAdd(&sumsq[c], b + s2[threadIdx.x + 128]);
  }
}

__global__ void bn_apply(const float* __restrict__ h, const float* __restrict__ sum,
                         const float* __restrict__ sumsq, const float* __restrict__ gamma,
                         const float* __restrict__ beta, float* __restrict__ xc,
                         int dst_off, int N) {
  size_t i = (size_t)blockIdx.x * blockDim.x + threadIdx.x;
  if (i >= (size_t)N * 128) return;
  int n = (int)(i >> 7), c = (int)(i & 127);
  float invN = 1.0f / (float)N;
  float mu = sum[c] * invN;
  float var = sumsq[c] * invN - mu * mu;
  float val = (h[i] - mu) * rsqrtf(var + 1e-5f) * gamma[c] + beta[c];
  xc[(size_t)n * 512 + dst_off + c] = val;
}

// ---------------------------------------------------------------- log softmax (C = 40)
__global__ void log_softmax40(const float* __restrict__ logits, float* __restrict__ out, int N) {
  int n = blockIdx.x * blockDim.x + threadIdx.x;
  if (n >= N) return;
  const float* l = logits + (size_t)n * 48;
  float m = l[0];
#pragma unroll
  for (int c = 1; c < 40; ++c) m = fmaxf(m, l[c]);
  float s = 0.f;
#pragma unroll
  for (int c = 0; c < 40; ++c) s += expf(l[c] - m);
  float ls = logf(s);
  float* o = out + (size_t)n * 40;
#pragma unroll
  for (int c = 0; c < 40; ++c) o[c] = l[c] - m - ls;
}

// ---------------------------------------------------------------- host launcher
extern "C" void kernel_launch(void* const* d_in, const int* in_sizes, int n_in,
                              void* d_out, int out_size, void* d_ws, size_t ws_size,
                              hipStream_t stream) {
  (void)n_in; (void)out_size; (void)ws_size;
  const float* x        = (const float*)d_in[0];
  const int*   ei       = (const int*)  d_in[1];
  const float* grid_mp  = (const float*)d_in[2];
  const float* cbw      = (const float*)d_in[3];
  const float* csw      = (const float*)d_in[4];
  const float* csc      = (const float*)d_in[5];
  const float* cbias    = (const float*)d_in[6];
  const float* bng      = (const float*)d_in[7];
  const float* bnb      = (const float*)d_in[8];
  const float* grid_out = (const float*)d_in[9];
  const float* obw      = (const float*)d_in[10];
  const float* osw      = (const float*)d_in[11];
  const float* osc      = (const float*)d_in[12];

  const int N = in_sizes[0] / F_IN;
  const int E = in_sizes[1] / 2;
  const int H = 128, C = 40, D = 512, OPAD = 48;

  // carve workspace
  uint8_t* p = (uint8_t*)d_ws;
  auto carve = [&](size_t bytes) -> void* {
    void* r = (void*)p; p += (bytes + 255) & ~(size_t)255; return r;
  };
  float* deg    = (float*)carve((size_t)N * 4);
  float* dinv   = (float*)carve((size_t)N * 4);
  float* xc     = (float*)carve((size_t)N * D * 4);
  float* hlin   = (float*)carve((size_t)N * H * 4);
  float* hagg   = (float*)carve((size_t)N * H * 4);
  float* Wc     = (float*)carve((size_t)KEXP * H * 4);
  float* Wo     = (float*)carve((size_t)OPAD * (D * 9) * 4);
  float* bsum   = (float*)carve(128 * 4);   // 512B-rounded -> bsq = bsum + 128
  float* bsq    = (float*)carve(128 * 4);
  float* logits = (float*)carve((size_t)N * OPAD * 4);

  auto cdiv = [](int a, int b) { return (a + b - 1) / b; };

  // degrees / dinv (identical every layer)
  fill_f32<<<cdiv(N, 256), 256, 0, stream>>>(deg, 0.0f, (size_t)N);
  deg_count<<<cdiv(E, 256), 256, 0, stream>>>(ei, deg, E);
  make_dinv<<<cdiv(N, 256), 256, 0, stream>>>(deg, dinv, N);

  // feats[0] = x into skip-concat slab
  copy_x_to_xc<<<cdiv(N * F_IN, 256), 256, 0, stream>>>(x, xc, N);

  for (int i = 0; i < 3; ++i) {
    prep_w<<<cdiv(H * F_IN, 256), 256, 0, stream>>>(
        cbw + (size_t)i * H * F_IN, csw + (size_t)i * H * F_IN * 8,
        csc + (size_t)i * H * F_IN, Wc, H, H, F_IN);

    dim3 gg(cdiv(N, 64), cdiv(H, 64));
    kan_gemm<<<gg, 256, 0, stream>>>(xc, D, i * 128, grid_mp, Wc, KEXP,
                                     hlin, H, N, H, 0);

    agg_init<<<cdiv(N * 128, 256), 256, 0, stream>>>(hlin, dinv, cbias + (size_t)i * H,
                                                     hagg, N);
    agg_edges<<<cdiv(E * 32, 256), 256, 0, stream>>>(ei, hlin, dinv, hagg, E);

    fill_f32<<<1, 256, 0, stream>>>(bsum, 0.0f, 256);   // bsum + bsq contiguous
    bn_reduce<<<cdiv(N, 128), 256, 0, stream>>>(hagg, bsum, bsq, N);
    bn_apply<<<cdiv(N * 128, 256), 256, 0, stream>>>(hagg, bsum, bsq,
                                                     bng + (size_t)i * H,
                                                     bnb + (size_t)i * H,
                                                     xc, (i + 1) * 128, N);
  }

  // output KANLinear over skip-concat (4 blocks of 128 features, accumulated)
  prep_w<<<cdiv(OPAD * D, 256), 256, 0, stream>>>(obw, osw, osc, Wo, C, OPAD, D);
  for (int b = 0; b < 4; ++b) {
    dim3 gg(cdiv(N, 64), 1);
    kan_gemm<<<gg, 256, 0, stream>>>(xc, D, b * 128,
                                     grid_out + (size_t)b * 128 * GPTS,
                                     Wo + (size_t)b * KEXP, D * 9,
                                     logits, OPAD, N, OPAD, b > 0);
  }

  log_softmax40<<<cdiv(N, 256), 256, 0, stream>>>(logits, (float*)d_out, N);
}